// PositionalEncoding3D_41308995453355
// MI455X (gfx1250) — compile-verified
//
#include <hip/hip_runtime.h>

// ---------------------------------------------------------------------------
// PositionalEncoding3D: out[b, d*256+h*16+w, c] =
//   c <  341 : pos_d[d][c]
//   c <  682 : pos_h[h][c-341]
//   c < 1023 : pos_w[w][c-682]
//   c == 1023: extra[d][h][w]
//
// 0-FLOP streaming op: 128 MiB stores, ~84 KB loads. Floor ~5.8us @ 23.3TB/s.
// Strategy per block (b,d,h): build the 16x1024 f32 slice (64 KB) in LDS with
// branch-free contiguous segment copies, then DMA it to HBM with a single
// TDM tensor_store_from_lds. pos_w staged in via tensor_load_to_lds,
// overlapped with the w-invariant segment build.
// gfx1250 paths: TDM load + store, s_wait_tensorcnt, wave32, LDS b128.
// ---------------------------------------------------------------------------

typedef float        v4f __attribute__((ext_vector_type(4)));
typedef unsigned int v4u __attribute__((ext_vector_type(4)));
typedef int          v4i __attribute__((ext_vector_type(4)));
typedef int          v8i __attribute__((ext_vector_type(8)));

#define DIM_D   16
#define DIM_H   16
#define DIM_W   16
#define CAX     341           // channels per axis
#define CMODEL  1024
#define BATCH   8

#if defined(__AMDGCN__) && __has_builtin(__builtin_amdgcn_tensor_load_to_lds) && \
    __has_builtin(__builtin_amdgcn_tensor_store_from_lds) &&                     \
    __has_builtin(__builtin_amdgcn_s_wait_tensorcnt)
#define USE_TDM 1
#else
#define USE_TDM 0
#endif

#if USE_TDM
// Build a 2D D# (ISA ch.8): tensor dim0 x dim1 elements of 4 bytes,
// tile == tensor, row stride == dim0 (fully contiguous transfer).
__device__ __forceinline__ void tdm_desc_2d(unsigned lds_off,
                                            unsigned long long ga,
                                            unsigned dim0, unsigned dim1,
                                            v4u* g0, v8i* g1) {
  v4u a;
  a[0] = 1u;                                           // count=1, user desc
  a[1] = lds_off;                                      // LDS byte address
  a[2] = (unsigned)(ga & 0xFFFFFFFFull);               // global_addr[31:0]
  a[3] = (unsigned)((ga >> 32) & 0x01FFFFFFull)        // global_addr[56:32]
         | 0x80000000u;                                // type=2 ("image")
  v8i b;
  b[0] = (int)(2u << 16);          // data_size code 2 -> 4 bytes
  b[1] = (int)(dim0 << 16);        // tensor_dim0[15:0]
  b[2] = (int)(dim1 << 16);        // tensor_dim0 hi=0 | tensor_dim1
  b[3] = (int)(dim0 << 16);        // tensor_dim1 hi=0 | tile_dim0
  b[4] = (int)dim1;                // tile_dim1 | tile_dim2=0
  b[5] = (int)dim0;                // tensor_dim0_stride[31:0]
  b[6] = 0;
  b[7] = 0;
  *g0 = a;
  *g1 = b;
}
#endif

__global__ __launch_bounds__(256) void pe3d_kernel(
    const float* __restrict__ pd,   // (16, 341)
    const float* __restrict__ ph,   // (16, 341)
    const float* __restrict__ pw,   // (16, 341)
    const float* __restrict__ ex,   // (16, 16, 16, 1)
    float* __restrict__ out)        // (8, 4096, 1024)
{
  __shared__ float pw_lds[DIM_W * CAX];      // 21,824 B: staged pos_w table
  __shared__ float slice[DIM_W * CMODEL];    // 65,536 B: 16 output rows

  const int tid = (int)threadIdx.x;
  const int blk = blockIdx.x;                // 0..2047 = (b, d, h)
  const int b = blk >> 8;
  const int d = (blk >> 4) & 15;
  const int h = blk & 15;

#if USE_TDM
  // Wave 0 kicks off the DMA of the full 16x341 pos_w table into LDS.
  // TDM ignores EXEC, so gate with a real (wave-uniform) branch.
  if (tid < 32) {
    v4u g0; v8i g1;
    tdm_desc_2d((unsigned)(unsigned long long)(void*)&pw_lds[0],
                (unsigned long long)(const void*)pw, CAX, DIM_W, &g0, &g1);
    const v4i gz4 = {0, 0, 0, 0};
    const v8i gz8 = {0, 0, 0, 0, 0, 0, 0, 0};
    __builtin_amdgcn_tensor_load_to_lds(g0, g1, gz4, gz4, gz8, 0);
  }
#else
  for (int i = tid; i < DIM_W * CAX; i += 256) pw_lds[i] = pw[i];
#endif

  // ---- Build w-invariant segments while the pos_w DMA is in flight. ----
  // seg0: slice[w][0..340]   = pos_d[d][:]   (replicated over w)
  // seg1: slice[w][341..681] = pos_h[h][:]   (replicated over w)
  for (int i = tid; i < CAX; i += 256) {
    const float vd = pd[d * CAX + i];
    const float vh = ph[h * CAX + i];
    #pragma unroll
    for (int w = 0; w < DIM_W; ++w) {
      slice[w * CMODEL + i]       = vd;
      slice[w * CMODEL + CAX + i] = vh;
    }
  }
  // seg3: slice[w][1023] = extra[d][h][w]
  if (tid < DIM_W) slice[tid * CMODEL + (CMODEL - 1)] = ex[(d * DIM_H + h) * DIM_W + tid];

#if USE_TDM
  if (tid < 32) __builtin_amdgcn_s_wait_tensorcnt(0);
#endif
  __syncthreads();

  // seg2: slice[w][682..1022] = pos_w[w][:]  (from LDS-staged table)
  for (int k = tid; k < DIM_W * CAX; k += 256) {
    const int w = k / CAX;            // const-div -> mul/shift
    const int j = k - w * CAX;
    slice[w * CMODEL + 2 * CAX + j] = pw_lds[k];
  }
  __syncthreads();

  // ---- Stream the finished 64 KB slice to HBM. ----
  const unsigned long long out_ga =
      (unsigned long long)(const void*)
      (out + ((size_t)(b * (DIM_D * DIM_H * DIM_W) + d * 256 + h * 16)) * CMODEL);

#if USE_TDM
  // One TDM DMA: 16 rows x 1024 ch, fully contiguous 64 KB store.
  if (tid < 32) {
    v4u g0; v8i g1;
    tdm_desc_2d((unsigned)(unsigned long long)(void*)&slice[0],
                out_ga, CMODEL, DIM_W, &g0, &g1);
    const v4i gz4 = {0, 0, 0, 0};
    const v8i gz8 = {0, 0, 0, 0, 0, 0, 0, 0};
    __builtin_amdgcn_tensor_store_from_lds(g0, g1, gz4, gz4, gz8, 0);
    __builtin_amdgcn_s_wait_tensorcnt(0);
  }
#else
  // Fallback: branch-free LDS b128 -> NT global b128 streaming loop.
  float* outp = (float*)out_ga;
  #pragma unroll
  for (int r = 0; r < (DIM_W * CMODEL) / (256 * 4); ++r) {  // 16 iters
    const int idx = (r * 256 + tid) * 4;
    const v4f v = *(const v4f*)&slice[idx];
    __builtin_nontemporal_store(v, (v4f*)(outp + idx));
  }
#endif
}

extern "C" void kernel_launch(void* const* d_in, const int* in_sizes, int n_in,
                              void* d_out, int out_size, void* d_ws, size_t ws_size,
                              hipStream_t stream) {
  // inputs: [0]=x (unused, defines B only), [1]=pos_d, [2]=pos_h, [3]=pos_w,
  //         [4]=pos_extra
  const float* pd = (const float*)d_in[1];
  const float* ph = (const float*)d_in[2];
  const float* pw = (const float*)d_in[3];
  const float* ex = (const float*)d_in[4];
  float* out = (float*)d_out;

  dim3 grid(BATCH * DIM_D * DIM_H);   // 2048 blocks: one 64 KB slice each
  dim3 block(256);                    // 8 wave32s
  hipLaunchKernelGGL(pe3d_kernel, grid, block, 0, stream, pd, ph, pw, ex, out);
}